// Multi_Heads_Self_Attn_Q_KV_1D_1812476199809
// MI455X (gfx1250) — compile-verified
//
#include <hip/hip_runtime.h>

// ---------------------------------------------------------------------------
// Softmax-free MHSA, algebraically collapsed (linear attention reassociation):
//   out[b] = X_q[b] + (gamma * Wo * Pt[b]) * X_q[b]
// where per batch:
//   S[b]   = X_kv[b] * X_kv[b]^T                      [256 x 256]   (K = 2048)
//   T1     = Wk_h * S[b]                              [64  x 256]
//   M_h    = (T1 * Wv_h^T) / 64                       [64  x 64 ]
//   Pt_h   = M_h^T * Wq_h                             [64  x 256]  (Pt[b] = P^T, [512 x 256] row-major)
//   G[b]   = gamma * Wo * Pt[b]                       [256 x 256]
//   out[b] = X_q[b] + G[b] * X_q[b]                   [256 x 2048]
// ~2.8 GFLOP total vs ~43 GFLOP naive; ~24 MB HBM traffic -> bandwidth bound
// (~1 us @ 23.3 TB/s), so fp32 WMMA (V_WMMA_F32_16X16X4_F32) end-to-end keeps
// reference-grade numerics at zero roofline cost.
// One wave32 computes a 16 x (16*NB) strip: A-fragments are reused across NB
// N-tiles (1.25 loads/WMMA at NB=4) and same-accumulator WMMAs are NB apart.
// ---------------------------------------------------------------------------

typedef float v2f __attribute__((ext_vector_type(2)));
typedef float v8f __attribute__((ext_vector_type(8)));

constexpr int B_   = 4;
constexpr int CQ_  = 256;
constexpr int CKV_ = 256;
constexpr int NQ_  = 2048;
constexpr int NKV_ = 2048;
constexpr int H_   = 8;
constexpr int DK_  = 64;
constexpr int OD_  = H_ * DK_;  // 512

// Generic strided WMMA GEMM strip: C(m0:m0+16, n0:n0+16*NB) = scale*(A.B) [+ R].
//   A element (m,k) at A[m*rsA + k*csA], B element (k,n) at B[k*rsB + n*csB].
// Fragment layouts per CDNA5 ISA 7.12.2 (wave32):
//   A 16x4 : lanes 0-15 -> M=lane, K=k0,k0+1 in v0,v1; lanes 16-31 -> K=k0+2,k0+3
//   B 4x16 : v_i holds row K=k0+i (lanes 0-15) / K=k0+2+i (lanes 16-31), N=lane&15
//   C/D    : VGPR v -> row M = v + 8*(lane>>4), col N = lane&15
template <int NB, bool HASR>
__device__ __forceinline__ void wmma_gemm_strip(
    const float* __restrict__ A, long rsA, long csA,
    const float* __restrict__ Bm, long rsB, long csB,
    float* __restrict__ C, long rsC,
    const float* __restrict__ R, long rsR,
    int m0, int n0, int K, float scale) {
  const int lane = threadIdx.x & 31;
  const int r    = lane & 15;
  const int half = lane >> 4;
  const long arow = (long)(m0 + r) * rsA;
  long bcol[NB];
#pragma unroll
  for (int j = 0; j < NB; ++j) bcol[j] = (long)(n0 + 16 * j + r) * csB;

  v8f acc[NB];
#pragma unroll
  for (int j = 0; j < NB; ++j) acc[j] = {};

  for (int k0 = 0; k0 < K; k0 += 8) {
    const int kA = k0 + 2 * half;
    // Unconditional near-scope prefetch of the streaming operand a few K-steps
    // ahead; speculative prefetch past the buffer end is silently dropped.
    __builtin_prefetch(&Bm[(long)(kA + 32) * rsB + bcol[0]], 0, 3);
    v2f a0, a1;
    a0.x = A[arow + (long)(kA    ) * csA];
    a0.y = A[arow + (long)(kA + 1) * csA];
    a1.x = A[arow + (long)(kA + 4) * csA];
    a1.y = A[arow + (long)(kA + 5) * csA];
    v2f b0[NB], b1[NB];
#pragma unroll
    for (int j = 0; j < NB; ++j) {
      b0[j].x = Bm[(long)(kA    ) * rsB + bcol[j]];
      b0[j].y = Bm[(long)(kA + 1) * rsB + bcol[j]];
      b1[j].x = Bm[(long)(kA + 4) * rsB + bcol[j]];
      b1[j].y = Bm[(long)(kA + 5) * rsB + bcol[j]];
    }
#pragma unroll
    for (int j = 0; j < NB; ++j)
      acc[j] = __builtin_amdgcn_wmma_f32_16x16x4_f32(false, a0, false, b0[j],
                                                     (short)0, acc[j], false, false);
#pragma unroll
    for (int j = 0; j < NB; ++j)
      acc[j] = __builtin_amdgcn_wmma_f32_16x16x4_f32(false, a1, false, b1[j],
                                                     (short)0, acc[j], false, false);
  }

#pragma unroll
  for (int j = 0; j < NB; ++j) {
#pragma unroll
    for (int v = 0; v < 8; ++v) {
      const long row = m0 + v + 8 * half;
      const long col = n0 + 16 * j + r;
      float val = acc[j][v] * scale;
      if (HASR) val += R[row * rsR + col];
      C[row * rsC + col] = val;
    }
  }
}

// Stage 1: S[b] = X_kv[b] * X_kv[b]^T     (M=N=256, K=2048)  NB=2 -> 512 waves
__global__ void k_stage1_xxT(const float* __restrict__ xkv, float* __restrict__ S) {
  const int b = blockIdx.z;
  const float* X = xkv + (long)b * CKV_ * NKV_;
  wmma_gemm_strip<2, false>(X, NKV_, 1, X, 1, NKV_,
                            S + (long)b * CKV_ * CKV_, CKV_, nullptr, 0,
                            blockIdx.y * 16, blockIdx.x * 32, NKV_, 1.0f);
}

// Stage 2: T1[b,h] = Wk_h * S[b]          (M=64, N=256, K=256)
__global__ void k_stage2_T1(const float* __restrict__ Wk, const float* __restrict__ S,
                            float* __restrict__ T1) {
  const int z = blockIdx.z;  // b*H + h
  const int b = z >> 3, h = z & 7;
  wmma_gemm_strip<4, false>(Wk + (long)h * DK_ * CKV_, CKV_, 1,
                            S + (long)b * CKV_ * CKV_, CKV_, 1,
                            T1 + (long)z * DK_ * CKV_, CKV_, nullptr, 0,
                            blockIdx.y * 16, blockIdx.x * 64, CKV_, 1.0f);
}

// Stage 3: M[b,h] = (T1[b,h] * Wv_h^T)/64 (M=64, N=64, K=256); B transposed via strides
__global__ void k_stage3_M(const float* __restrict__ T1, const float* __restrict__ Wv,
                           float* __restrict__ Mm) {
  const int z = blockIdx.z;
  const int h = z & 7;
  wmma_gemm_strip<4, false>(T1 + (long)z * DK_ * CKV_, CKV_, 1,
                            Wv + (long)h * DK_ * CKV_, 1, CKV_,  // B(k,n)=Wv_h[n][k]
                            Mm + (long)z * DK_ * DK_, DK_, nullptr, 0,
                            blockIdx.y * 16, blockIdx.x * 64, CKV_, 1.0f / (float)DK_);
}

// Stage 4: Pt[b,h] = M[b,h]^T * Wq_h      (M=64, N=256, K=64); A transposed via strides
__global__ void k_stage4_Pt(const float* __restrict__ Mm, const float* __restrict__ Wq,
                            float* __restrict__ Pt) {
  const int z = blockIdx.z;
  const int h = z & 7;
  wmma_gemm_strip<4, false>(Mm + (long)z * DK_ * DK_, 1, DK_,  // A(m,k)=M[k][m]
                            Wq + (long)h * DK_ * CQ_, CQ_, 1,
                            Pt + (long)z * DK_ * CQ_, CQ_, nullptr, 0,
                            blockIdx.y * 16, blockIdx.x * 64, DK_, 1.0f);
}

// Stage 5: G[b] = gamma * Wo * Pt[b]      (M=256, N=256, K=512)
__global__ void k_stage5_G(const float* __restrict__ Wo, const float* __restrict__ Pt,
                           const float* __restrict__ gamma, float* __restrict__ G) {
  const int b = blockIdx.z;
  wmma_gemm_strip<4, false>(Wo, OD_, 1,
                            Pt + (long)b * OD_ * CQ_, CQ_, 1,
                            G + (long)b * CQ_ * CQ_, CQ_, nullptr, 0,
                            blockIdx.y * 16, blockIdx.x * 64, OD_, gamma[0]);
}

// Stage 6: out[b] = X_q[b] + G[b]*X_q[b]  (M=256, N=2048, K=256), residual epilogue
__global__ void k_stage6_out(const float* __restrict__ G, const float* __restrict__ xq,
                             float* __restrict__ out) {
  const int b = blockIdx.z;
  const float* Bq = xq + (long)b * CQ_ * NQ_;
  wmma_gemm_strip<4, true>(G + (long)b * CQ_ * CQ_, CQ_, 1,
                           Bq, NQ_, 1,
                           out + (long)b * CQ_ * NQ_, NQ_, Bq, NQ_,
                           blockIdx.y * 16, blockIdx.x * 64, CQ_, 1.0f);
}

extern "C" void kernel_launch(void* const* d_in, const int* in_sizes, int n_in,
                              void* d_out, int out_size, void* d_ws, size_t ws_size,
                              hipStream_t stream) {
  const float* xq    = (const float*)d_in[0];
  const float* xkv   = (const float*)d_in[1];
  const float* Wq    = (const float*)d_in[2];
  const float* Wk    = (const float*)d_in[3];
  const float* Wv    = (const float*)d_in[4];
  const float* Wo    = (const float*)d_in[5];
  const float* gamma = (const float*)d_in[6];
  float* out = (float*)d_out;

  // Workspace carve-up (floats): 6.8 MB total, all L2-resident.
  float* S  = (float*)d_ws;                       // [B][256][256]
  float* T1 = S  + (size_t)B_ * CKV_ * CKV_;      // [B][H][64][256]
  float* Mm = T1 + (size_t)B_ * H_ * DK_ * CKV_;  // [B][H][64][64]
  float* Pt = Mm + (size_t)B_ * H_ * DK_ * DK_;   // [B][H][64][256]
  float* G  = Pt + (size_t)B_ * H_ * DK_ * CQ_;   // [B][256][256]

  const dim3 blk(32, 1, 1);  // exactly one wave32 per output strip
  k_stage1_xxT<<<dim3(CKV_ / 32, CKV_ / 16, B_),     blk, 0, stream>>>(xkv, S);
  k_stage2_T1 <<<dim3(CKV_ / 64, DK_ / 16, B_ * H_), blk, 0, stream>>>(Wk, S, T1);
  k_stage3_M  <<<dim3(DK_ / 64,  DK_ / 16, B_ * H_), blk, 0, stream>>>(T1, Wv, Mm);
  k_stage4_Pt <<<dim3(CQ_ / 64,  DK_ / 16, B_ * H_), blk, 0, stream>>>(Mm, Wq, Pt);
  k_stage5_G  <<<dim3(CQ_ / 64,  CQ_ / 16, B_),      blk, 0, stream>>>(Wo, Pt, gamma, G);
  k_stage6_out<<<dim3(NQ_ / 64,  CQ_ / 16, B_),      blk, 0, stream>>>(G, xq, out);
}